// SimpleScan_48790828482568
// MI455X (gfx1250) — compile-verified
//
#include <hip/hip_runtime.h>
#include <hip/hip_bf16.h>

typedef float v2f __attribute__((ext_vector_type(2)));
typedef float v4f __attribute__((ext_vector_type(4)));
typedef float v8f __attribute__((ext_vector_type(8)));

#define BATCH 128
#define SEQT  1024
#define OUTN  6
#define KMAX  416   // max padded K (layer 3: 208+208)

__device__ __forceinline__ float sigf(float x) { return 1.0f / (1.0f + __expf(-x)); }

// ---------------------------------------------------------------------------
// Repack [Wx ; Wh] (row-major [K,4H]) into padded, per-lane WMMA-B fragment
// order, packed as float4 per lane covering a k-pair (-> global_load_b128).
// Packed float index:
//   idx = ((tile*kpairs + kp)*32 + lane)*4 + j      (j = 0..3)
//   kk  = 2*kp + (j>>1)                             (k-step)
//   k   = 4*kk + 2*(lane>>4) + (j&1)                (B frag: v0=K{0,2}, v1=K{1,3})
//   np  = tile*16 + (lane&15)                       (padded gate column; gates
//                                                    i,f,g,o each padded to Hp)
// Zero padding rows/cols so padded math is exact.
// ---------------------------------------------------------------------------
__global__ void repack_kernel(const float* __restrict__ Wx,
                              const float* __restrict__ Wh,
                              const float* __restrict__ b,
                              float* __restrict__ Wp, float* __restrict__ bp,
                              int Din, int DinP, int H, int Hp, int Kp)
{
    const int  fourH  = 4 * H;
    const int  kpairs = Kp >> 3;             // ksteps/2
    const long long total = 4LL * Hp * Kp;   // (4Hp/16)*(Kp/8)*128

    for (long long idx = (long long)blockIdx.x * blockDim.x + threadIdx.x;
         idx < total;
         idx += (long long)gridDim.x * blockDim.x) {
        int j    = (int)(idx & 3);
        int lane = (int)((idx >> 2) & 31);
        long long rest = idx >> 7;
        int kp   = (int)(rest % kpairs);
        int tile = (int)(rest / kpairs);
        int kk   = 2 * kp + (j >> 1);
        int k    = 4 * kk + 2 * (lane >> 4) + (j & 1);
        int np   = tile * 16 + (lane & 15);
        int gate = np / Hp;
        int n    = np - gate * Hp;
        float v  = 0.0f;
        if (n < H) {
            int col = gate * H + n;
            if (k < DinP) {
                if (k < Din) v = Wx[(long long)k * fourH + col];
            } else {
                int kh = k - DinP;
                if (kh < H) v = Wh[(long long)kh * fourH + col];
            }
        }
        Wp[idx] = v;
    }

    long long tid = (long long)blockIdx.x * blockDim.x + threadIdx.x;
    if (tid < 4LL * Hp) {
        int gate = (int)tid / Hp, n = (int)tid - gate * Hp;
        bp[tid] = (n < H) ? b[gate * H + n] : 0.0f;
    }
}

// ---------------------------------------------------------------------------
// One LSTM layer time scan.  grid.x = B/16 batch groups; one wave per 16-wide
// hidden tile (blockDim = (Hp/16)*32).  A = [x_t || h_{t-1}] lives in LDS;
// cell state c lives in WMMA C-layout registers for all 1024 steps.
// Inner loop per k-pair: 1x ds_load_2addr_b64 + 4x global_load_b128 + 8x WMMA.
// ---------------------------------------------------------------------------
__global__ void lstm_scan_kernel(const float* __restrict__ X,   // [B, T, Xs]
                                 const float* __restrict__ Wp,  // packed frags
                                 const float* __restrict__ bp,  // [4*Hp]
                                 float* __restrict__ Hseq,      // [B, T, Hp]
                                 int Xs, int DinP, int Hp, int Kp)
{
    __shared__ float a_buf[16 * KMAX];           // [m][k], stride Kp

    const int lane   = threadIdx.x & 31;
    const int wave   = threadIdx.x >> 5;         // hidden-tile index
    const int col    = lane & 15;                // N col within tile / A row
    const int half   = lane >> 4;
    const int kpairs = Kp >> 3;
    const int blockB = blockIdx.x * 16;
    const int htiles = Hp >> 4;

    // zero whole A buffer (h region must start at 0; x region overwritten)
    for (int i = threadIdx.x; i < 16 * Kp; i += blockDim.x) a_buf[i] = 0.0f;

    // per-lane bias for this wave's gate columns
    const int nglob = wave * 16 + col;
    const float bi = bp[0 * Hp + nglob];
    const float bf = bp[1 * Hp + nglob];
    const float bg = bp[2 * Hp + nglob];
    const float bo = bp[3 * Hp + nglob];

    // packed-B fragment base pointers for the four gate tiles of this wave
    const v4f* __restrict__ Wv = (const v4f*)Wp;
    const v4f* __restrict__ pI = Wv + ((long long)(0 * htiles + wave)) * kpairs * 32 + lane;
    const v4f* __restrict__ pF = Wv + ((long long)(1 * htiles + wave)) * kpairs * 32 + lane;
    const v4f* __restrict__ pG = Wv + ((long long)(2 * htiles + wave)) * kpairs * 32 + lane;
    const v4f* __restrict__ pO = Wv + ((long long)(3 * htiles + wave)) * kpairs * 32 + lane;

    const float* __restrict__ aRow = a_buf + col * Kp + 2 * half;   // A frag base

    v8f c_reg = {0.f, 0.f, 0.f, 0.f, 0.f, 0.f, 0.f, 0.f};
    const v8f vzero = {0.f, 0.f, 0.f, 0.f, 0.f, 0.f, 0.f, 0.f};

    __syncthreads();

    for (int t = 0; t < SEQT; ++t) {
        // stage x_t into the A buffer (coalesced, cooperative)
        for (int i = threadIdx.x; i < 16 * DinP; i += blockDim.x) {
            int m = i / DinP;
            int k = i - m * DinP;
            a_buf[m * Kp + k] = X[((long long)(blockB + m) * SEQT + t) * Xs + k];
        }
        __syncthreads();

        v8f acc_i = vzero, acc_f = vzero, acc_g = vzero, acc_o = vzero;

        // z = [x_t || h] @ Wp  — four independent WMMA chains per wave
        for (int kp = 0; kp < kpairs; ++kp) {
            v2f a0 = *(const v2f*)(aRow + (kp << 3));
            v2f a1 = *(const v2f*)(aRow + (kp << 3) + 4);
            v4f wi = pI[(long long)kp << 5];
            v4f wf = pF[(long long)kp << 5];
            v4f wg = pG[(long long)kp << 5];
            v4f wo = pO[(long long)kp << 5];
            acc_i = __builtin_amdgcn_wmma_f32_16x16x4_f32(false, a0, false,
                        __builtin_shufflevector(wi, wi, 0, 1), (short)0, acc_i, false, false);
            acc_f = __builtin_amdgcn_wmma_f32_16x16x4_f32(false, a0, false,
                        __builtin_shufflevector(wf, wf, 0, 1), (short)0, acc_f, false, false);
            acc_g = __builtin_amdgcn_wmma_f32_16x16x4_f32(false, a0, false,
                        __builtin_shufflevector(wg, wg, 0, 1), (short)0, acc_g, false, false);
            acc_o = __builtin_amdgcn_wmma_f32_16x16x4_f32(false, a0, false,
                        __builtin_shufflevector(wo, wo, 0, 1), (short)0, acc_o, false, false);
            acc_i = __builtin_amdgcn_wmma_f32_16x16x4_f32(false, a1, false,
                        __builtin_shufflevector(wi, wi, 2, 3), (short)0, acc_i, false, false);
            acc_f = __builtin_amdgcn_wmma_f32_16x16x4_f32(false, a1, false,
                        __builtin_shufflevector(wf, wf, 2, 3), (short)0, acc_f, false, false);
            acc_g = __builtin_amdgcn_wmma_f32_16x16x4_f32(false, a1, false,
                        __builtin_shufflevector(wg, wg, 2, 3), (short)0, acc_g, false, false);
            acc_o = __builtin_amdgcn_wmma_f32_16x16x4_f32(false, a1, false,
                        __builtin_shufflevector(wo, wo, 2, 3), (short)0, acc_o, false, false);
        }
        __syncthreads();   // all reads of h_{t-1} done before h_t overwrite

        // gates + state update; C layout: row m = r + 8*half, col = nglob
        #pragma unroll
        for (int r = 0; r < 8; ++r) {
            float zi = acc_i[r] + bi;
            float zf = acc_f[r] + bf;
            float zg = acc_g[r] + bg;
            float zo = acc_o[r] + bo;
            float c  = sigf(zf) * c_reg[r] + sigf(zi) * tanhf(zg);
            c_reg[r] = c;
            float h  = sigf(zo) * tanhf(c);
            int m = r + 8 * half;
            a_buf[m * Kp + DinP + nglob] = h;
            Hseq[((long long)(blockB + m) * SEQT + t) * Hp + nglob] = h;
        }
        __syncthreads();   // h_t visible before next iteration reads it
    }
}

// ---------------------------------------------------------------------------
// out[b,o] = bd[o] + sum_n h_last[b,n] * Wd[n,o]    (tiny: 128x6, K=200)
// ---------------------------------------------------------------------------
__global__ void dense_out_kernel(const float* __restrict__ Hseq,  // [B,T,Hp]
                                 const float* __restrict__ Wd,    // [H,OUT]
                                 const float* __restrict__ bd,
                                 float* __restrict__ out,
                                 int Hp, int H)
{
    int idx = blockIdx.x * blockDim.x + threadIdx.x;
    if (idx >= BATCH * OUTN) return;
    int b = idx / OUTN, o = idx - b * OUTN;
    const float* h = Hseq + ((long long)b * SEQT + (SEQT - 1)) * Hp;
    float s = bd[o];
    for (int n = 0; n < H; ++n) s += h[n] * Wd[n * OUTN + o];
    out[idx] = s;
}

// ---------------------------------------------------------------------------
extern "C" void kernel_launch(void* const* d_in, const int* in_sizes, int n_in,
                              void* d_out, int out_size, void* d_ws, size_t ws_size,
                              hipStream_t stream)
{
    (void)in_sizes; (void)n_in; (void)out_size; (void)ws_size;

    const float* xs = (const float*)d_in[0];
    const float* Wx[4] = {(const float*)d_in[1], (const float*)d_in[4],
                          (const float*)d_in[7], (const float*)d_in[10]};
    const float* Wh[4] = {(const float*)d_in[2], (const float*)d_in[5],
                          (const float*)d_in[8], (const float*)d_in[11]};
    const float* bb[4] = {(const float*)d_in[3], (const float*)d_in[6],
                          (const float*)d_in[9], (const float*)d_in[12]};
    const float* Wd = (const float*)d_in[13];
    const float* bd = (const float*)d_in[14];
    float* out = (float*)d_out;

    // layer geometry: Din, DinP(padded in), H, Hp(padded H), Kp = DinP+Hp
    const int Din [4] = {128, 100, 100, 200};
    const int DinP[4] = {128, 112, 112, 208};
    const int H   [4] = {100, 100, 200, 200};
    const int Hp  [4] = {112, 112, 208, 208};
    const int Kp  [4] = {240, 224, 320, 416};

    // workspace layout (floats)
    float* ws = (float*)d_ws;
    size_t off = 0;
    float* Wp[4]; float* bp[4];
    for (int l = 0; l < 4; ++l) { Wp[l] = ws + off; off += (size_t)4 * Hp[l] * Kp[l]; }
    for (int l = 0; l < 4; ++l) { bp[l] = ws + off; off += (size_t)4 * Hp[l]; }
    const size_t seqElems = (size_t)BATCH * SEQT * 208;   // max-stride buffer
    float* seqA = ws + off; off += seqElems;
    float* seqB = ws + off; off += seqElems;

    // 1) repack weights + bias into WMMA fragment order (zero-padded)
    for (int l = 0; l < 4; ++l) {
        long long total = 4LL * Hp[l] * Kp[l];
        int blocks = (int)((total + 255) / 256);
        repack_kernel<<<blocks, 256, 0, stream>>>(Wx[l], Wh[l], bb[l],
                                                  Wp[l], bp[l],
                                                  Din[l], DinP[l], H[l], Hp[l], Kp[l]);
    }

    // 2) four layer scans, ping-ponging sequence buffers
    const float* Xin[4]  = {xs,   seqA, seqB, seqA};
    float*       Hout[4] = {seqA, seqB, seqA, seqB};
    const int    Xs[4]   = {128, 112, 112, 208};
    for (int l = 0; l < 4; ++l) {
        dim3 grid(BATCH / 16);
        dim3 block((Hp[l] / 16) * 32);
        lstm_scan_kernel<<<grid, block, 0, stream>>>(Xin[l], Wp[l], bp[l], Hout[l],
                                                     Xs[l], DinP[l], Hp[l], Kp[l]);
    }

    // 3) dense head from last timestep of layer 3 (in seqB)
    dense_out_kernel<<<(BATCH * OUTN + 127) / 128, 128, 0, stream>>>(
        seqB, Wd, bd, out, Hp[3], H[3]);
}